// DiscriminatorAugment_15702400434793
// MI455X (gfx1250) — compile-verified
//
#include <hip/hip_runtime.h>

// ---------------------------------------------------------------------------
// DiscriminatorAugment for MI455X (gfx1250, wave32).
// Bandwidth-bound: ~600 MB total traffic -> ~26us roofline at 23.3 TB/s.
// Pass 1 (sum of translated image) uses V_WMMA_F32_16X16X4_F32 with a ones
// B-matrix as a 64-floats-per-instruction reduction on the matrix pipe.
// Pass 2 fuses translate+brightness+saturation+contrast+cutout with NT stores.
// ---------------------------------------------------------------------------

typedef float v2f __attribute__((ext_vector_type(2)));
typedef float v4f __attribute__((ext_vector_type(4)));
typedef float v8f __attribute__((ext_vector_type(8)));

#define Bk 64
#define Ck 3
#define Hk 512
#define Wk 512
#define NPIX (Hk * Wk)          // 262144 pixels per channel
#define NPB  (Ck * NPIX)        // 786432 elements per batch image
#define NBLK_SUM 64             // sum-pass blocks per batch
#define CHUNK (NPB / NBLK_SUM)  // 12288 elements per block
#define WAVE_CHUNK (CHUNK / 8)  // 1536 elements per wave (8 waves/block)
#define ITERS (WAVE_CHUNK / 64) // 24 wmma iterations per wave (64 elem each)

// Workspace float-index layout (wsf = float view, wsi = int view, same buffer):
//   [0..63]     brightness (r2-0.5)
//   [64..127]   saturation scale (r3*2)
//   [128..191]  contrast scale (r4+0.5)
//   [192..255]  g_mean of translated image (written by aug_reduce)
//   [256..319]  th   [320..383] tw
//   [384..447]  cutout row0  [448..511] row1  [512..575] col0  [576..639] col1
//   [1024..1024+64*64) per-block partial sums

__global__ void aug_setup(const float* __restrict__ rand01,
                          float* __restrict__ wsf, int* __restrict__ wsi) {
  int b = threadIdx.x;
  if (b >= Bk) return;
  float r0 = rand01[0 * Bk + b];
  float r1 = rand01[1 * Bk + b];
  float r2 = rand01[2 * Bk + b];
  float r3 = rand01[3 * Bk + b];
  float r4 = rand01[4 * Bk + b];
  float r5 = rand01[5 * Bk + b];
  float r6 = rand01[6 * Bk + b];
  // delta_h = delta_w = round(512*0.125) = 64 ; th,tw in [-64,64]
  int th = (int)floorf(r0 * 129.0f) - 64;
  int tw = (int)floorf(r1 * 129.0f) - 64;
  // cutout: ch=cw=round(512*0.2)=102 (even) -> oh,ow = floor(r*513), box = [o-51, o+50] clipped
  int oh = (int)floorf(r5 * 513.0f);
  int ow = (int)floorf(r6 * 513.0f);
  wsf[b]        = r2 - 0.5f;
  wsf[64 + b]   = r3 * 2.0f;
  wsf[128 + b]  = r4 + 0.5f;
  wsi[256 + b]  = th;
  wsi[320 + b]  = tw;
  wsi[384 + b]  = max(0, oh - 51);
  wsi[448 + b]  = min(Hk - 1, oh + 50);
  wsi[512 + b]  = max(0, ow - 51);
  wsi[576 + b]  = min(Wk - 1, ow + 50);
}

// Pass 1: per-block partial sums of the translated (zero-padded-gather) image.
// Each wave accumulates via WMMA f32 16x16x4 with an all-ones B matrix:
// every wmma folds 64 freshly loaded floats into the 16x16 accumulator.
// Loads are clamp+select (branch-free) so EXEC is all-ones at the WMMA.
__global__ void aug_sum(const float* __restrict__ img,
                        const int* __restrict__ wsi,
                        float* __restrict__ partial) {
  int b    = blockIdx.y;
  int th   = wsi[256 + b];
  int tw   = wsi[320 + b];
  int wave = threadIdx.x >> 5;
  int lane = threadIdx.x & 31;
  int base = blockIdx.x * CHUNK + wave * WAVE_CHUNK + lane * 2;
  const float* src = img + (long)b * NPB;

  v8f acc = {};
  v2f ones; ones.x = 1.0f; ones.y = 1.0f;

#pragma unroll 4
  for (int it = 0; it < ITERS; ++it) {
    int idx = base + it * 64;               // two consecutive elements idx, idx+1
    int c   = idx >> 18;                    // / (512*512)
    int rem = idx & (NPIX - 1);
    int y   = rem >> 9;
    int x   = rem & (Wk - 1);               // x is even; x+1 stays in the same row
    int sy  = y + th;
    int sx  = x + tw;
    bool okY = (sy >= 0) && (sy < Hk);
    int syc  = min(max(sy, 0), Hk - 1);
    int sx0  = min(max(sx, 0), Wk - 1);
    int sx1  = min(max(sx + 1, 0), Wk - 1);
    const float* row = src + ((long)c << 18) + ((long)syc << 9);
    float f0 = (okY && sx     >= 0 && sx     < Wk) ? 1.0f : 0.0f;
    float f1 = (okY && sx + 1 >= 0 && sx + 1 < Wk) ? 1.0f : 0.0f;
    v2f a;
    a.x = row[sx0] * f0;
    a.y = row[sx1] * f1;
    // D[i][j] = sum_k A[i,k] * 1 ; arrangement of loaded values in A is
    // irrelevant since everything is summed with the ones matrix.
    acc = __builtin_amdgcn_wmma_f32_16x16x4_f32(
        false, a, false, ones, (short)0, acc, false, false);
  }

  // acc[r] in lane j holds C[M][j]: lanes 0-15 -> M=r, lanes 16-31 -> M=8+r.
  // C[i][j] is the same for all j, so total = (lane0 VGPR-sum) + (lane16 VGPR-sum).
  float s = acc[0] + acc[1] + acc[2] + acc[3] + acc[4] + acc[5] + acc[6] + acc[7];
  float o = __shfl_down(s, 16, 32);
  float wave_tot = s + o;                   // valid in lane 0

  __shared__ float sm[8];
  if (lane == 0) sm[wave] = wave_tot;
  __syncthreads();
  if (threadIdx.x == 0) {
    float t = 0.0f;
#pragma unroll
    for (int i = 0; i < 8; ++i) t += sm[i]; // fixed order -> deterministic
    partial[b * NBLK_SUM + blockIdx.x] = t;
  }
}

// Pass 1b: fold 64 partials per batch into g_mean (deterministic LDS tree).
__global__ void aug_reduce(const float* __restrict__ partial,
                           float* __restrict__ wsf) {
  __shared__ float sm[64];
  int b = blockIdx.x;
  int t = threadIdx.x;
  sm[t] = partial[b * NBLK_SUM + t];
  __syncthreads();
#pragma unroll
  for (int off = 32; off > 0; off >>= 1) {
    if (t < off) sm[t] += sm[t + off];
    __syncthreads();
  }
  if (t == 0) wsf[192 + b] = sm[0] * (1.0f / (float)NPB);
}

// Pass 2: fused translate + brightness + saturation + contrast + cutout.
// One thread = 4 consecutive pixels across all 3 channels.
// Input re-read mostly hits the 192MB L2; output uses NT stores so the
// streamed writes don't evict the input between passes.
__global__ void aug_apply(const float* __restrict__ img,
                          const float* __restrict__ wsf,
                          const int* __restrict__ wsi,
                          float* __restrict__ out) {
  int b = blockIdx.y;
  float br = wsf[b];
  float s2 = wsf[64 + b];
  float s3 = wsf[128 + b];
  // reference g_mean is taken after brightness; saturation preserves it:
  float g  = wsf[192 + b] + br;
  int th = wsi[256 + b], tw = wsi[320 + b];
  int r0 = wsi[384 + b], r1 = wsi[448 + b];
  int c0 = wsi[512 + b], c1 = wsi[576 + b];

  int p  = (blockIdx.x * blockDim.x + threadIdx.x) * 4;  // pixel index in [0, NPIX)
  int y  = p >> 9;
  int x0 = p & (Wk - 1);
  int sy = y + th;
  bool okY = (sy >= 0) && (sy < Hk);
  int syc  = min(max(sy, 0), Hk - 1);
  const float* src = img + (long)b * NPB + ((long)syc << 9);

  v4f o0, o1, o2;

#pragma unroll
  for (int j = 0; j < 4; ++j) {
    int xx  = x0 + j;
    int sx  = xx + tw;
    int sxc = min(max(sx, 0), Wk - 1);
    float f = (okY && sx >= 0 && sx < Wk) ? 1.0f : 0.0f;
    float v0 = src[0 * NPIX + sxc] * f + br;
    float v1 = src[1 * NPIX + sxc] * f + br;
    float v2 = src[2 * NPIX + sxc] * f + br;
    float chm = (v0 + v1 + v2) * (1.0f / 3.0f);
    v0 = (v0 - chm) * s2 + chm;
    v1 = (v1 - chm) * s2 + chm;
    v2 = (v2 - chm) * s2 + chm;
    v0 = (v0 - g) * s3 + g;
    v1 = (v1 - g) * s3 + g;
    v2 = (v2 - g) * s3 + g;
    float m = (y >= r0 && y <= r1 && xx >= c0 && xx <= c1) ? 0.0f : 1.0f;
    o0[j] = v0 * m;
    o1[j] = v1 * m;
    o2[j] = v2 * m;
  }

  float* ob = out + (long)b * NPB + p;
  __builtin_nontemporal_store(o0, (v4f*)(ob));
  __builtin_nontemporal_store(o1, (v4f*)(ob + NPIX));
  __builtin_nontemporal_store(o2, (v4f*)(ob + 2 * NPIX));
}

extern "C" void kernel_launch(void* const* d_in, const int* in_sizes, int n_in,
                              void* d_out, int out_size, void* d_ws, size_t ws_size,
                              hipStream_t stream) {
  const float* images = (const float*)d_in[0];   // (64,3,512,512) f32
  const float* rand01 = (const float*)d_in[1];   // (7,64,1,1) f32
  float* out = (float*)d_out;
  float* wsf = (float*)d_ws;
  int*   wsi = (int*)d_ws;
  float* partial = wsf + 1024;

  aug_setup<<<1, 64, 0, stream>>>(rand01, wsf, wsi);
  aug_sum<<<dim3(NBLK_SUM, Bk), 256, 0, stream>>>(images, wsi, partial);
  aug_reduce<<<Bk, 64, 0, stream>>>(partial, wsf);
  aug_apply<<<dim3(NPIX / 1024, Bk), 256, 0, stream>>>(images, wsf, wsi, out);
}